// DamagedPointRepair_6571299963098
// MI455X (gfx1250) — compile-verified
//
#include <hip/hip_runtime.h>
#include <stdint.h>

// ---------------------------------------------------------------------------
// DamagedPointRepair on MI455X (gfx1250)
//
// Bandwidth-bound stencil: min traffic = 8192^2 * 8 B = 512 MB -> ~22 us at
// 23.3 TB/s. Compute ~1.3 GFLOP (negligible) => WMMA is irrelevant; the CDNA5
// leverage is the data-movement path:
//   * Tensor Data Mover: ONE tensor_load_to_lds per block stages the 66x66
//     halo tile into LDS (iterated 1-row descriptor: lds stride 66, global
//     stride 8192), tracked by TENSORcnt / s_wait_tensorcnt.
//   * Zero-padded SAME-conv halo: DMA window clamped to the image; border
//     blocks pre-zero only the uncovered LDS ring (disjoint from DMA region).
//   * Non-temporal streaming stores (256 MB image > 192 MB L2).
//   * VALU kept under the memory floor: interior blocks (97%) take a
//     division-free fast path (cnt==4 -> *0.25f exact); rolling-row stencil
//     cuts LDS reads from 9 to 3 per pixel.
// ---------------------------------------------------------------------------

typedef __attribute__((ext_vector_type(4))) uint32_t u32x4;
typedef __attribute__((ext_vector_type(8))) uint32_t u32x8;

#define W_IMG 8192
#define H_IMG 8192
#define TILE  64
#define LW    66   // TILE + 2 halo

__global__ __launch_bounds__(256)
void DamagedPointRepair_kernel(const float* __restrict__ img,
                               float* __restrict__ out) {
  __shared__ float sm[LW * LW];

  const int col0 = blockIdx.x * TILE;
  const int row0 = blockIdx.y * TILE;

  // Clamped halo window [sy,ey) x [sx,ex) in the global image.
  const int gx0 = col0 - 1, gy0 = row0 - 1;
  const int sx = gx0 < 0 ? 0 : gx0;
  const int sy = gy0 < 0 ? 0 : gy0;
  const int ex = (gx0 + LW) > W_IMG ? W_IMG : (gx0 + LW);
  const int ey = (gy0 + LW) > H_IMG ? H_IMG : (gy0 + LW);
  const int xoff = sx - gx0;   // 0 or 1
  const int yoff = sy - gy0;   // 0 or 1
  const int vw = ex - sx;      // valid width  (<= 66)
  const int vh = ey - sy;      // valid height (<= 66)

  const bool border = (blockIdx.x == 0) | (blockIdx.y == 0) |
                      (blockIdx.x == gridDim.x - 1) | (blockIdx.y == gridDim.y - 1);

  // Border blocks: zero the LDS cells the DMA will NOT write (zero padding).
  if (border) {
    for (int idx = threadIdx.x; idx < LW * LW; idx += 256) {
      const int ly = idx / LW;
      const int lx = idx - ly * LW;
      const int gy = gy0 + ly;
      const int gx = gx0 + lx;
      if ((unsigned)gx >= (unsigned)W_IMG || (unsigned)gy >= (unsigned)H_IMG)
        sm[idx] = 0.0f;
    }
  }

  // ---- One TDM DMA per workgroup (wave 0 issues; TENSORcnt is per-wave) ----
  if ((threadIdx.x >> 5) == 0) {
    const uint64_t gaddr = (uint64_t)(uintptr_t)img +
                           ((uint64_t)sy * W_IMG + (uint64_t)sx) * 4u;
    // Low 32 bits of a generic pointer into __shared__ = LDS byte offset.
    const uint32_t laddr = (uint32_t)(uintptr_t)(&sm[yoff * LW + xoff]);

    // D# group 0: count=1 | lds_addr | global_addr[56:0] | type=2
    u32x4 g0;
    g0.x = 1u;
    g0.y = laddr;
    g0.z = (uint32_t)gaddr;
    g0.w = ((uint32_t)(gaddr >> 32) & 0x01FFFFFFu) | (2u << 30);

    // D# group 1: data_size=4B (code 2), iterate_enable=1; tile = vw x 1 row.
    u32x8 g1;
    g1.s0 = (2u << 16) | (1u << 19);
    g1.s1 = ((uint32_t)vw & 0xFFFFu) << 16;                              // dim0[15:0]
    g1.s2 = ((uint32_t)vw >> 16) | (((uint32_t)vh & 0xFFFFu) << 16);     // dim0 hi | dim1 lo
    g1.s3 = ((uint32_t)vh >> 16) | ((uint32_t)vw << 16);                 // dim1 hi | tile_dim0
    g1.s4 = 1u;                                                          // tile_dim1=1
    g1.s5 = (uint32_t)W_IMG;                                             // dim0_stride lo32
    g1.s6 = 0u;
    g1.s7 = 0u;

    // D# group 2 (iterate mode): lds_inc=66 elems, global_inc=8192 elems,
    // iterate_count = vh-1  (vh rows total).
    u32x4 g2;
    g2.x = 0u;
    g2.y = (uint32_t)LW;
    g2.z = (uint32_t)W_IMG;
    g2.w = ((uint32_t)(vh - 1) & 0xFFFFu) << 16;

    u32x4 g3 = {0u, 0u, 0u, 0u};

    asm volatile("tensor_load_to_lds %0, %1, %2, %3"
                 :
                 : "s"(g0), "s"(g1), "s"(g2), "s"(g3)
                 : "memory");
#if __has_builtin(__builtin_amdgcn_s_wait_tensorcnt)
    __builtin_amdgcn_s_wait_tensorcnt(0);
#else
    asm volatile("s_wait_tensorcnt 0x0" ::: "memory");
#endif
  }
  __syncthreads();

  // ---- Compute: rolling-row stencil, 16 consecutive rows per thread ----
  const int tx = threadIdx.x & 63;        // column within tile
  const int ty = threadIdx.x >> 6;        // 0..3 -> rows [ty*16, ty*16+16)
  const int gc = col0 + tx;
  const int rbase = ty * 16;
  const float* base = &sm[rbase * LW + tx];   // top-left of first 3x3 window
  float* orow = &out[(size_t)(row0 + rbase) * W_IMG + gc];

  // Prime rows A (above) and B (center) of the first window.
  float a1  = base[1];
  float rsA = (base[0] + a1) + base[2];
  float b0  = base[LW], b1 = base[LW + 1], b2 = base[LW + 2];
  float rsB = (b0 + b1) + b2;

  if (!border) {
    // -------- interior fast path: coeff == 1, cnt == 4 (exact *0.25f) ------
#pragma unroll
    for (int i = 0; i < 16; ++i) {
      const float* rp = base + (size_t)(i + 2) * LW;
      const float c0 = rp[0], c1 = rp[1], c2 = rp[2];
      const float rsC = (c0 + c1) + c2;

      const float sum9 = (rsA + rsB) + rsC;
      const float mean = sum9 * (1.0f / 9.0f);
      const bool  bad  = (b1 > 5.0f * mean) | (b1 > 1000.0f);

      const float nsum = (a1 + c1) + (b0 + b2);
      const float rep  = __builtin_floorf(nsum * 0.25f);

      __builtin_nontemporal_store(bad ? rep : b1, orow);
      orow += W_IMG;

      a1 = b1; rsA = rsB;
      b0 = c0; b1 = c1; b2 = c2; rsB = rsC;
    }
  } else {
    // -------- general path: edge coefficients + exact floor(nsum/cnt) ------
    const bool  ec   = (gc == 0) | (gc == W_IMG - 1);
    const float ccnt = (float)((gc > 0) + (gc < W_IMG - 1));
#pragma unroll
    for (int i = 0; i < 16; ++i) {
      const float* rp = base + (size_t)(i + 2) * LW;
      const float c0 = rp[0], c1 = rp[1], c2 = rp[2];
      const float rsC = (c0 + c1) + c2;

      const int   gr    = row0 + rbase + i;
      const float sum9  = (rsA + rsB) + rsC;
      const bool  er    = (gr == 0) | (gr == H_IMG - 1);
      const float coeff = (er & ec) ? 2.25f : ((er | ec) ? 1.5f : 1.0f);
      const float mean  = sum9 * (1.0f / 9.0f) * coeff;
      const bool  bad   = (b1 > 5.0f * mean) | (b1 > 1000.0f);

      const float nsum = (a1 + c1) + (b0 + b2);
      const float cnt  = (float)((gr > 0) + (gr < H_IMG - 1)) + ccnt;
      const float rep  = __builtin_floorf(nsum / cnt);

      __builtin_nontemporal_store(bad ? rep : b1, orow);
      orow += W_IMG;

      a1 = b1; rsA = rsB;
      b0 = c0; b1 = c1; b2 = c2; rsB = rsC;
    }
  }
}

extern "C" void kernel_launch(void* const* d_in, const int* in_sizes, int n_in,
                              void* d_out, int out_size, void* d_ws, size_t ws_size,
                              hipStream_t stream) {
  (void)in_sizes; (void)n_in; (void)out_size; (void)d_ws; (void)ws_size;
  const float* img = (const float*)d_in[0];
  float* out = (float*)d_out;
  dim3 grid(W_IMG / TILE, H_IMG / TILE);   // 128 x 128 blocks
  DamagedPointRepair_kernel<<<grid, dim3(256), 0, stream>>>(img, out);
}